// AdaptiveMetaLearnerV2_39453569581470
// MI455X (gfx1250) — compile-verified
//
#include <hip/hip_runtime.h>

typedef __attribute__((ext_vector_type(2))) float v2f;
typedef __attribute__((ext_vector_type(8))) float v8f;

#define NT 4                 // N tiles: 64 cols (60 live: n = 3k + {i,g,o}, f gate is dead)
#define KT 5                 // K tiles: K = 20 = 5*4, exact fit for wmma f32 16x16x4
#define WAVES 8
#define TILES_PER_WAVE 16
#define ROWS_PER_BLOCK (WAVES * TILES_PER_WAVE * 16)   // 2048
#define PDIM 32768
#define BDIM 64
#define GSTRIDE 20           // LDS col stride in floats: 80B => 16B-aligned b128 stores

__device__ __forceinline__ float fast_tanh(float x) {
#if __has_builtin(__builtin_amdgcn_tanhf)
  return __builtin_amdgcn_tanhf(x);
#else
  // CDNA5 trans op V_TANH_F32; pad for the trans-op result hazard
  float r;
  asm volatile("v_tanh_f32 %0, %1\n\tv_nop\n\tv_nop" : "=v"(r) : "v"(x));
  return r;
#endif
}

__device__ __forceinline__ v8f wmma4(v2f a, v2f b, v8f c) {
  // D = A(16x4,f32) * B(4x16,f32) + C(16x16,f32)
  return __builtin_amdgcn_wmma_f32_16x16x4_f32(
      /*neg_a=*/false, a, /*neg_b=*/false, b,
      /*c_mod=*/(short)0, c, /*reuse_a=*/false, /*reuse_b=*/false);
}

__device__ __forceinline__ float sigm(float x) {
  return fmaf(fast_tanh(0.5f * x), 0.5f, 0.5f);
}

__global__ __launch_bounds__(256) void meta_lstm_kernel(
    const float* __restrict__ x,    const float* __restrict__ W1,
    const float* __restrict__ b1,   const float* __restrict__ Wih,
    const float* __restrict__ bih,  const float* __restrict__ bhh,
    const float* __restrict__ Wout, const float* __restrict__ bout,
    const float* __restrict__ Wact, const float* __restrict__ bact,
    float* __restrict__ theta, float* __restrict__ blockq) {
  __shared__ float gbuf[WAVES][64 * GSTRIDE];
  __shared__ float wq[WAVES];

  const int tid  = threadIdx.x;
  const int wave = tid >> 5;
  const int lane = tid & 31;
  const int lm   = lane & 15;   // row (A) or column (B/D) within tile
  const int half = lane >> 4;   // K-half selector per WMMA layout

  // ---------- preload per-lane weight fragments (live across all tiles) ----------
  // B fragment layout: VGPR v, lane j -> B[k0 + v + 2*(j>>4), j&15]
  float2 Bw[2][NT][KT];
  float  Cb[2][NT];
#pragma unroll
  for (int l = 0; l < 2; ++l) {
#pragma unroll
    for (int nt = 0; nt < NT; ++nt) {
      const int cc = nt * 16 + lm;       // permuted gate column
      const bool valid = cc < 60;
      const int k = valid ? (cc / 3) : 0;
      const int g = valid ? (cc % 3) : 0;
      const int grow = (g == 0 ? 0 : (g == 1 ? 40 : 60)) + k;  // i,g,o rows of W_ih
      Cb[l][nt] = valid ? (bih[l * 80 + grow] + bhh[l * 80 + grow]) : 0.0f;
#pragma unroll
      for (int kt = 0; kt < KT; ++kt) {
        const int kk = kt * 4 + 2 * half;
        float2 wv = make_float2(0.0f, 0.0f);
        if (valid) wv = *(const float2*)(Wih + (size_t)l * 1600 + grow * 20 + kk);
        Bw[l][nt][kt] = wv;
      }
    }
  }
  // linear1 per-lane constants: inp[m,k] = x[m]*W1[k] + b1[k]
  float2 w1p[KT], b1p[KT];
#pragma unroll
  for (int kt = 0; kt < KT; ++kt) {
    const int kk = kt * 4 + 2 * half;
    w1p[kt] = make_float2(W1[kk], W1[kk + 1]);
    b1p[kt] = make_float2(b1[kk], b1[kk + 1]);
  }
  // fused output head: col0 = W_out, col1 = W_act, rest zero
  float2 Bo[KT];
#pragma unroll
  for (int kt = 0; kt < KT; ++kt) {
    const int kk = kt * 4 + 2 * half;
    float2 wv = make_float2(0.0f, 0.0f);
    if (lm == 0)      wv = make_float2(Wout[kk], Wout[kk + 1]);
    else if (lm == 1) wv = make_float2(Wact[kk], Wact[kk + 1]);
    Bo[kt] = wv;
  }
  float Cbo = (lm == 0) ? bout[0] : ((lm == 1) ? bact[0] : 0.0f);

  float* gb = gbuf[wave];
  float qacc = 0.0f;
  const long rowbase0 = (long)blockIdx.x * ROWS_PER_BLOCK + wave * (TILES_PER_WAVE * 16);

  for (int t = 0; t < TILES_PER_WAVE; ++t) {
    const long rowbase = rowbase0 + (long)t * 16;
    const float xv = x[rowbase + lm];

    // layer-1 input as A fragments: lane -> inp[lm, kt*4 + 2*half + {0,1}]
    v2f A[KT];
#pragma unroll
    for (int kt = 0; kt < KT; ++kt) {
      A[kt][0] = fmaf(xv, w1p[kt].x, b1p[kt].x);
      A[kt][1] = fmaf(xv, w1p[kt].y, b1p[kt].y);
    }

#pragma unroll
    for (int l = 0; l < 2; ++l) {
      // gates(16x64) = A(16x20) x B(20x64) + bias  (20 WMMAs)
#pragma unroll
      for (int nt = 0; nt < NT; ++nt) {
        const float bias = Cb[l][nt];
        v8f acc = {bias, bias, bias, bias, bias, bias, bias, bias};
#pragma unroll
        for (int kt = 0; kt < KT; ++kt) {
          v2f bf; bf[0] = Bw[l][nt][kt].x; bf[1] = Bw[l][nt][kt].y;
          acc = wmma4(A[kt], bf, acc);
        }
        // D -> LDS, column-major [c][m]: 8 consecutive rows per lane => 2x b128
        float4* p = (float4*)(gb + (nt * 16 + lm) * GSTRIDE + half * 8);
        p[0] = make_float4(acc[0], acc[1], acc[2], acc[3]);
        p[1] = make_float4(acc[4], acc[5], acc[6], acc[7]);
      }
      // activations: h = sigm(o) * tanh(sigm(i) * tanh(g)); f gate dead (c0 = 0).
      // Each h consumed exactly once, directly in next layer's A-fragment slot.
#pragma unroll
      for (int kt = 0; kt < KT; ++kt) {
        const int kk = kt * 4 + 2 * half;
#pragma unroll
        for (int j = 0; j < 2; ++j) {
          const float* q = gb + 3 * (kk + j) * GSTRIDE + lm;
          const float gi = q[0];
          const float gg = q[GSTRIDE];
          const float go = q[2 * GSTRIDE];
          const float ct = fast_tanh(sigm(gi) * fast_tanh(gg));
          A[kt][j] = sigm(go) * ct;
        }
      }
    }

    // output head: [theta | act] = h(16x20) x Bo(20x2)  (5 WMMAs)
    v8f accO = {Cbo, Cbo, Cbo, Cbo, Cbo, Cbo, Cbo, Cbo};
#pragma unroll
    for (int kt = 0; kt < KT; ++kt) {
      v2f bf; bf[0] = Bo[kt].x; bf[1] = Bo[kt].y;
      accO = wmma4(A[kt], bf, accO);
    }
    if (lm == 0) {  // lanes 0 & 16 hold theta rows 0-7 / 8-15: 64B coalesced
      float4* o = (float4*)(theta + rowbase + half * 8);
      o[0] = make_float4(accO[0], accO[1], accO[2], accO[3]);
      o[1] = make_float4(accO[4], accO[5], accO[6], accO[7]);
    } else if (lm == 1) {  // lanes 1 & 17 hold act rows: accumulate qt partial
      qacc += (accO[0] + accO[1]) + (accO[2] + accO[3]) +
              ((accO[4] + accO[5]) + (accO[6] + accO[7]));
    }
  }

  // deterministic qt reduction: lanes 1,17 -> wave -> block partial in d_ws
  const float q = __shfl(qacc, 1, 32) + __shfl(qacc, 17, 32);
  if (lane == 0) wq[wave] = q;
  __syncthreads();
  if (tid == 0) {
    float s = 0.0f;
#pragma unroll
    for (int w = 0; w < WAVES; ++w) s += wq[w];
    blockq[blockIdx.x] = s;
  }
}

__global__ void qt_reduce_kernel(const float* __restrict__ blockq,
                                 float* __restrict__ qt) {
  const int b = threadIdx.x;
  if (b < BDIM) {
    const int BPB = PDIM / ROWS_PER_BLOCK;  // 16 blocks per batch row
    float s = 0.0f;
    for (int j = 0; j < BPB; ++j) s += blockq[b * BPB + j];
    qt[b] = s * (1.0f / (float)PDIM);
  }
}

extern "C" void kernel_launch(void* const* d_in, const int* in_sizes, int n_in,
                              void* d_out, int out_size, void* d_ws, size_t ws_size,
                              hipStream_t stream) {
  (void)in_sizes; (void)n_in; (void)out_size; (void)ws_size;
  const float* x    = (const float*)d_in[0];
  const float* W1   = (const float*)d_in[1];
  const float* b1   = (const float*)d_in[2];
  const float* Wih  = (const float*)d_in[3];
  const float* bih  = (const float*)d_in[4];
  // d_in[5] = W_hh: dead (h0 = 0)
  const float* bhh  = (const float*)d_in[6];
  const float* Wout = (const float*)d_in[7];
  const float* bout = (const float*)d_in[8];
  const float* Wact = (const float*)d_in[9];
  const float* bact = (const float*)d_in[10];
  float* out    = (float*)d_out;
  float* blockq = (float*)d_ws;

  const int BP = BDIM * PDIM;                 // 2,097,152 rows
  dim3 grid(BP / ROWS_PER_BLOCK);             // 1024 blocks, 16 per batch row
  meta_lstm_kernel<<<grid, 256, 0, stream>>>(x, W1, b1, Wih, bih, bhh,
                                             Wout, bout, Wact, bact,
                                             out, blockq);
  qt_reduce_kernel<<<1, 64, 0, stream>>>(blockq, out + BP);
}